// BboxNMS_31129922962028
// MI455X (gfx1250) — compile-verified
//
#include <hip/hip_runtime.h>

// ---------------------------------------------------------------------------
// Batched class-aware NMS for MI455X (gfx1250, wave32).
//   B=16 batches, N=2048 boxes, 21 classes, thresh=0.3.
// One workgroup (1024 threads = 32 waves) per batch; all per-batch state lives
// in LDS (320KB/WGP budget, we use ~57KB). Serial greedy NMS is restructured
// as 64 chunks of 32 boxes: intra-chunk suppression runs on ONE wave32 using
// __shfl broadcast + __ballot (exact fit for wave32), inter-chunk suppression
// is fully parallel with LDS ds_and_b32 atomics on the keep bitmask.
// Score staging uses gfx1250 async global->LDS (ASYNCcnt) when available.
// ---------------------------------------------------------------------------

typedef __attribute__((ext_vector_type(4))) float    v4f;
typedef __attribute__((ext_vector_type(4))) unsigned v4u;

// Builtin prototype (from clang diagnostic): first param is
//   'int __attribute__((vector_size(16))) __device__ *'  (AS1, non-const)
typedef int v4i_as __attribute__((vector_size(16)));
typedef __attribute__((address_space(1))) v4i_as* gptr_v4i;
typedef __attribute__((address_space(3))) v4i_as* lptr_v4i;

#define NMS_THRES 0.3f
#define NB 16
#define NN 2048
#define NCLS 21
#define BLOCK 1024

#if __has_builtin(__builtin_amdgcn_global_load_async_to_lds_b128)
#define HAS_ASYNC_LDS 1
#else
#define HAS_ASYNC_LDS 0
#endif

__device__ __forceinline__ void lds_async_copy16(const void* g, void* l) {
#if HAS_ASYNC_LDS
  // global_load_async_to_lds_b128: per-lane 16B global -> LDS, ASYNCcnt-tracked
  __builtin_amdgcn_global_load_async_to_lds_b128(
      (gptr_v4i)g, (lptr_v4i)l, /*offset=*/0, /*cpol=*/0);
#else
  *(v4u*)l = *(const v4u*)g;
#endif
}

__device__ __forceinline__ void lds_async_wait0() {
#if HAS_ASYNC_LDS
#if __has_builtin(__builtin_amdgcn_s_wait_asynccnt)
  __builtin_amdgcn_s_wait_asynccnt(0);
#else
  asm volatile("s_wait_asynccnt 0" ::: "memory");
#endif
#endif
}

// ---------------------------------------------------------------------------
// Kernel 1: streaming pass-through of encoded_cls / encoded_reg (NT hints).
// ---------------------------------------------------------------------------
__global__ void bboxnms_copy_kernel(const v4f* __restrict__ a, int na,
                                    const v4f* __restrict__ b, int nb,
                                    v4f* __restrict__ oa, v4f* __restrict__ ob) {
  int n = na + nb;
  for (int i = blockIdx.x * blockDim.x + threadIdx.x; i < n;
       i += gridDim.x * blockDim.x) {
    if (i < na) {
      v4f v = __builtin_nontemporal_load(a + i);
      __builtin_nontemporal_store(v, oa + i);
    } else {
      v4f v = __builtin_nontemporal_load(b + (i - na));
      __builtin_nontemporal_store(v, ob + (i - na));
    }
  }
}

// ---------------------------------------------------------------------------
// Kernel 2: per-batch NMS. grid = B blocks, block = 1024 threads (32 waves).
// ---------------------------------------------------------------------------
__global__ __launch_bounds__(BLOCK) void bboxnms_nms_kernel(
    const float* __restrict__ boxes,    // [B,N,4] xywh
    const float* __restrict__ scores,   // [B,N]
    const int*   __restrict__ classes,  // [B,N]
    float* __restrict__ out_boxes,      // [B,N,4]
    float* __restrict__ out_scores,     // [B,N]
    float* __restrict__ out_classes,    // [B,N]
    float* __restrict__ out_keep) {     // [B,N]
  __shared__ alignas(16) float skey[NN];     // sorted scores (keys)
  __shared__ int      sidx[NN];              // sorted original indices
  __shared__ v4f      sbox[NN];              // class-offset xyxy, sorted order
  __shared__ float    sarea[NN];             // areas, sorted order
  __shared__ unsigned keepmask[NN / 32];     // 2048-bit keep mask
  __shared__ v4f      cbox[32];              // current chunk boxes
  __shared__ float    carea[32];             // current chunk areas
  __shared__ unsigned curmask;               // surviving chunk members
  __shared__ float    redv[BLOCK / 32];      // cross-wave max reduction
  __shared__ float    maxc_s;

  const int b   = blockIdx.x;
  const int tid = threadIdx.x;
  const float* gb = boxes   + (size_t)b * NN * 4;
  const float* gs = scores  + (size_t)b * NN;
  const int*   gc = classes + (size_t)b * NN;

  // --- stage scores into LDS via async b128 copies (ASYNCcnt path) ---------
  for (int t = tid; t < NN / 4; t += BLOCK)
    lds_async_copy16(gs + 4 * t, &skey[4 * t]);

  // init index payload + keep mask while the async copies are in flight
  for (int i = tid; i < NN; i += BLOCK) sidx[i] = i;
  if (tid < NN / 32) keepmask[tid] = 0xFFFFFFFFu;

  // --- max coordinate over all xyxy entries (for the class-offset trick) ---
  float lm = -1e30f;
  for (int i = tid; i < NN; i += BLOCK) {
    v4f rb = ((const v4f*)gb)[i];  // x, y, w, h
    float x2 = rb.x + rb.z, y2 = rb.y + rb.w;
    lm = fmaxf(lm, fmaxf(fmaxf(rb.x, rb.y), fmaxf(x2, y2)));
  }
#pragma unroll
  for (int o = 16; o > 0; o >>= 1) lm = fmaxf(lm, __shfl_xor(lm, o, 32));
  if ((tid & 31) == 0) redv[tid >> 5] = lm;

  lds_async_wait0();   // scores resident in LDS for this wave
  __syncthreads();     // ... and for every wave

  if (tid < BLOCK / 32) {
    float v = redv[tid];
#pragma unroll
    for (int o = 16; o > 0; o >>= 1) v = fmaxf(v, __shfl_xor(v, o, 32));
    if (tid == 0) maxc_s = v;
  }
  __syncthreads();
  const float clsoff = maxc_s + 1.0f;

  // --- bitonic sort (descending by score), payload = original index --------
  for (int k = 2; k <= NN; k <<= 1) {
    for (int j = k >> 1; j > 0; j >>= 1) {
      __syncthreads();
      for (int i = tid; i < NN; i += BLOCK) {
        int ixj = i ^ j;
        if (ixj > i) {
          bool desc = ((i & k) == 0);
          float a = skey[i], c = skey[ixj];
          bool sw = desc ? (a < c) : (a > c);
          if (sw) {
            skey[i] = c; skey[ixj] = a;
            int t = sidx[i]; sidx[i] = sidx[ixj]; sidx[ixj] = t;
          }
        }
      }
    }
  }
  __syncthreads();

  // --- build class-offset xyxy boxes + areas in sorted order ---------------
  for (int i = tid; i < NN; i += BLOCK) {
    int s = sidx[i];
    v4f rb = ((const v4f*)gb)[s];                 // x, y, w, h
    float off = (float)gc[s] * clsoff;
    v4f ob;
    ob.x = rb.x + off;
    ob.y = rb.y + off;
    ob.z = rb.x + rb.z + off;
    ob.w = rb.y + rb.w + off;
    sbox[i]  = ob;
    sarea[i] = rb.z * rb.w;                       // offsets cancel in area
  }
  __syncthreads();

  // --- greedy NMS, 64 chunks of 32 (one wave32 per intra-chunk pass) -------
  for (int c = 0; c < NN / 32; ++c) {
    const int base = c * 32;

    // Phase 1: wave 0 resolves suppression among the 32 chunk boxes.
    if (tid < 32) {
      v4f   bi = sbox[base + tid];
      float ai = sarea[base + tid];
      cbox[tid]  = bi;
      carea[tid] = ai;
      unsigned m = keepmask[c];                   // uniform across the wave
#pragma unroll 1
      for (int p = 0; p < 31; ++p) {
        if ((m >> p) & 1u) {
          float x1 = __shfl(bi.x, p, 32);
          float y1 = __shfl(bi.y, p, 32);
          float x2 = __shfl(bi.z, p, 32);
          float y2 = __shfl(bi.w, p, 32);
          float ap = __shfl(ai,   p, 32);
          float iw = fmaxf(fminf(x2, bi.z) - fmaxf(x1, bi.x), 0.0f);
          float ih = fmaxf(fminf(y2, bi.w) - fmaxf(y1, bi.y), 0.0f);
          float inter = iw * ih;
          bool sup = (tid > p) && ((m >> tid) & 1u) &&
                     (inter > NMS_THRES * (ap + ai - inter));
          m &= ~((unsigned)__ballot(sup));        // wave32: low 32 bits
        }
      }
      if (tid == 0) { keepmask[c] = m; curmask = m; }
    }
    __syncthreads();

    // Phase 2: all threads suppress j > chunk_end against surviving members.
    const unsigned cm = curmask;
    const int jstart = base + 32;
    if (cm && jstart < NN) {
      for (int j = jstart + tid; j < NN; j += BLOCK) {
        unsigned w = keepmask[j >> 5];
        if (!((w >> (j & 31)) & 1u)) continue;    // already dead
        v4f   bj = sbox[j];
        float aj = sarea[j];
        bool dead = false;
        unsigned mm = cm;
        while (mm) {
          int t = __ffs(mm) - 1;
          mm &= mm - 1;
          v4f bb = cbox[t];
          float iw = fmaxf(fminf(bb.z, bj.z) - fmaxf(bb.x, bj.x), 0.0f);
          float ih = fmaxf(fminf(bb.w, bj.w) - fmaxf(bb.y, bj.y), 0.0f);
          float inter = iw * ih;
          if (inter > NMS_THRES * (carea[t] + aj - inter)) { dead = true; break; }
        }
        if (dead) atomicAnd(&keepmask[j >> 5], ~(1u << (j & 31)));  // ds_and_b32
      }
    }
    __syncthreads();
  }

  // --- outputs: xyxy->xywh round-trips to the raw box, so gather raw -------
  for (int i = tid; i < NN; i += BLOCK) {
    bool kp = (keepmask[i >> 5] >> (i & 31)) & 1u;
    int  s  = sidx[i];
    v4f  rb = ((const v4f*)gb)[s];
    v4f  ob = kp ? rb : v4f{0.0f, 0.0f, 0.0f, 0.0f};
    ((v4f*)out_boxes)[(size_t)b * NN + i] = ob;
    out_scores [(size_t)b * NN + i] = kp ? skey[i] : 0.0f;
    out_classes[(size_t)b * NN + i] = kp ? (float)gc[s] : -1.0f;
    out_keep   [(size_t)b * NN + i] = kp ? 1.0f : 0.0f;
  }
}

// ---------------------------------------------------------------------------
extern "C" void kernel_launch(void* const* d_in, const int* in_sizes, int n_in,
                              void* d_out, int out_size, void* d_ws,
                              size_t ws_size, hipStream_t stream) {
  const float* enc_cls = (const float*)d_in[0];  // [B,N,21]
  const float* enc_reg = (const float*)d_in[1];  // [B,N,4]
  const float* boxes   = (const float*)d_in[2];  // [B,N,4] xywh
  const float* scores  = (const float*)d_in[3];  // [B,N]
  const int*   classes = (const int*)d_in[4];    // [B,N]
  float* out = (float*)d_out;

  const size_t off_reg = (size_t)NB * NN * NCLS;           // 688128
  const size_t off_box = off_reg + (size_t)NB * NN * 4;    // 819200
  const size_t off_sc  = off_box + (size_t)NB * NN * 4;    // 950272
  const size_t off_cl  = off_sc  + (size_t)NB * NN;        // 983040
  const size_t off_kp  = off_cl  + (size_t)NB * NN;        // 1015808

  const int na = (int)(off_reg / 4);            // 172032 x 16B
  const int nb = (int)((size_t)NB * NN);        //  32768 x 16B

  bboxnms_copy_kernel<<<dim3(512), dim3(256), 0, stream>>>(
      (const v4f*)enc_cls, na, (const v4f*)enc_reg, nb,
      (v4f*)out, (v4f*)(out + off_reg));

  bboxnms_nms_kernel<<<dim3(NB), dim3(BLOCK), 0, stream>>>(
      boxes, scores, classes,
      out + off_box, out + off_sc, out + off_cl, out + off_kp);
}